// RGCNBlock_29377576305387
// MI455X (gfx1250) — compile-verified
//
#include <hip/hip_runtime.h>

typedef float v2f __attribute__((ext_vector_type(2)));
typedef float v8f __attribute__((ext_vector_type(8)));
typedef unsigned int v4u __attribute__((ext_vector_type(4)));
typedef int v4i __attribute__((ext_vector_type(4)));
typedef int v8i __attribute__((ext_vector_type(8)));

#define N_NODES 100000
#define DDIM    256
#define R_REL   5
#define E_EDGES 200000

#define KCHUNK  32                       // K rows of W staged per TDM transfer
#define NCHUNK  (DDIM / KCHUNK)          // 8
#define BUFELEM (KCHUNK * DDIM)          // 8192 floats = 32 KB per buffer

#if defined(__has_builtin)
#  if __has_builtin(__builtin_amdgcn_tensor_load_to_lds)
#    define RGCN_HAVE_TDM 1
#  endif
#endif
#ifndef RGCN_HAVE_TDM
#  define RGCN_HAVE_TDM 0
#endif

// ---------------------------------------------------------------------------
// Scatter: A[dst,:] += x[src,:] ; deg[dst] += 1   (f32 atomics, L2-resident A)
// ---------------------------------------------------------------------------
__global__ void rgcn_scatter_kernel(const float* __restrict__ x,
                                    const int* __restrict__ src,
                                    const int* __restrict__ dst,
                                    float* __restrict__ A,
                                    float* __restrict__ deg) {
    long long idx = (long long)blockIdx.x * blockDim.x + threadIdx.x;
    int e = (int)(idx >> 6);          // edge index
    int g = (int)(idx & 63);          // 4-dim group
    if (e >= E_EDGES) return;
    int s = src[e];
    int d = dst[e];
    int c0 = g * 4;
    const float4 v = *(const float4*)(x + (long long)s * DDIM + c0);
    float* ap = A + (long long)d * DDIM + c0;
    atomicAdd(ap + 0, v.x);
    atomicAdd(ap + 1, v.y);
    atomicAdd(ap + 2, v.z);
    atomicAdd(ap + 3, v.w);
    if (g == 0) atomicAdd(deg + d, 1.0f);
}

// ---------------------------------------------------------------------------
__device__ __forceinline__ v8f wmma_f32(v2f a, v2f b, v8f c) {
    return __builtin_amdgcn_wmma_f32_16x16x4_f32(
        /*neg_a=*/false, a, /*neg_b=*/false, b,
        /*c_mod=*/(short)0, c, /*reuse_a=*/false, /*reuse_b=*/false);
}

// ---------------------------------------------------------------------------
// out[row, col] += (A[row,:] * 1/max(deg[row],1)) @ W
//
// blockDim = 128 (4 waves). Block tile = 32 rows x 256 cols; each wave owns a
// 32x64 tile (8 x v8f accumulators) so every B operand feeds two WMMAs.
// W is staged HBM -> LDS in 32x256 chunks by the Tensor Data Mover
// (double-buffered, issued by wave 0, tracked with TENSORcnt), overlapping
// DMA with V_WMMA_F32_16X16X4_F32 compute on the previous chunk.
// ---------------------------------------------------------------------------
__global__ void rgcn_gemm_acc_kernel(const float* __restrict__ A,
                                     const float* __restrict__ deg,   // null => scale 1
                                     const float* __restrict__ W,     // (256,256) row major
                                     float* __restrict__ out) {
    extern __shared__ float smem[];                 // 2 x 32KB W chunk buffers
    const int tid  = threadIdx.x;
    const int lane = tid & 31;
    const int wv   = tid >> 5;                      // 0..3 -> column group
    const int m    = lane & 15;
    const int half = lane >> 4;                     // 0/1 -> K+0 / K+2
    const int rowBase = blockIdx.x * 32;
    const int colBase = wv * 64;

    const int aRow0 = rowBase + m;
    const int aRow1 = rowBase + 16 + m;
    float s0 = 1.0f, s1 = 1.0f;
    if (deg != nullptr) {
        s0 = 1.0f / fmaxf(deg[aRow0], 1.0f);
        s1 = 1.0f / fmaxf(deg[aRow1], 1.0f);
    }
    const float* __restrict__ ar0 = A + (long long)aRow0 * DDIM;
    const float* __restrict__ ar1 = A + (long long)aRow1 * DDIM;

    v8f c[4][2];
#pragma unroll
    for (int j = 0; j < 4; ++j) {
        c[j][0] = (v8f){0.f,0.f,0.f,0.f,0.f,0.f,0.f,0.f};
        c[j][1] = (v8f){0.f,0.f,0.f,0.f,0.f,0.f,0.f,0.f};
    }

#if RGCN_HAVE_TDM
    const unsigned int ldsBase = (unsigned int)(uintptr_t)smem;
#endif

    // ---- stage one 32x256 chunk of W into LDS buffer (kc & 1) ----
    auto stage = [&](int kc) {
#if RGCN_HAVE_TDM
        if (wv == 0) {                               // TDM ignores EXEC: issue once per block
            unsigned long long ga =
                (unsigned long long)(uintptr_t)(W + (size_t)kc * KCHUNK * DDIM);
            v4u g0;
            g0.x = 1u;                                        // count=1, user descriptor
            g0.y = ldsBase + (unsigned int)((kc & 1) * BUFELEM * 4);  // lds_addr (bytes)
            g0.z = (unsigned int)(ga & 0xFFFFFFFFu);          // global_addr[31:0]
            g0.w = (unsigned int)((ga >> 32) & 0x01FFFFFFu)   // global_addr[56:32]
                 | 0x80000000u;                               // type=2 ("image")
            v8i g1;
            g1[0] = 0x00020000;                // wg_mask=0, data_size=2 (4B)
            g1[1] = (int)(DDIM   << 16);       // tensor_dim0 = 256
            g1[2] = (int)(KCHUNK << 16);       // tensor_dim1 = 32
            g1[3] = (int)(DDIM   << 16);       // tile_dim0   = 256
            g1[4] = KCHUNK;                    // tile_dim1   = 32
            g1[5] = DDIM;                      // tensor_dim0_stride = 256
            g1[6] = 0;
            g1[7] = 0;
            v4i gz4 = (v4i){0, 0, 0, 0};       // 2-D tensor: groups 2/3 unused
            v8i gz8 = (v8i){0, 0, 0, 0, 0, 0, 0, 0};
            __builtin_amdgcn_tensor_load_to_lds(g0, g1, gz4, gz4, gz8, 0);
        }
#else
        const float* __restrict__ src = W + (size_t)kc * KCHUNK * DDIM;
        float* dstb = smem + (kc & 1) * BUFELEM;
        for (int i = tid * 4; i < BUFELEM; i += 128 * 4) {
            *(float4*)(dstb + i) = *(const float4*)(src + i);
        }
#endif
    };
    auto stage_wait = [&]() {
#if RGCN_HAVE_TDM
        if (wv == 0) __builtin_amdgcn_s_wait_tensorcnt(0);
#endif
        __syncthreads();
    };

    stage(0);
    stage_wait();

    for (int kc = 0; kc < NCHUNK; ++kc) {
        if (kc + 1 < NCHUNK) stage(kc + 1);          // overlap next DMA with compute

        const int bufOff = (kc & 1) * BUFELEM;
        const int kgBase = kc * KCHUNK;
#pragma unroll 4
        for (int k0 = 0; k0 < KCHUNK; k0 += 4) {
            const int ka = k0 + half * 2;            // chunk-local K for this lane half
            const int kg = kgBase + ka;              // global K
            v2f a0, a1;
            a0.x = ar0[kg]     * s0;
            a0.y = ar0[kg + 1] * s0;
            a1.x = ar1[kg]     * s1;
            a1.y = ar1[kg + 1] * s1;
            const int wb = bufOff + ka * DDIM + colBase + m;
#pragma unroll
            for (int j = 0; j < 4; ++j) {
                v2f b;
                b.x = smem[wb + j * 16];
                b.y = smem[wb + j * 16 + DDIM];
                c[j][0] = wmma_f32(a0, b, c[j][0]);  // rows rowBase..+15
                c[j][1] = wmma_f32(a1, b, c[j][1]);  // rows rowBase+16..+31
            }
        }

        __syncthreads();                             // all waves done with buf (kc&1)
        if (kc + 1 < NCHUNK) stage_wait();           // next chunk landed in LDS
    }

    // Accumulate the 32x64 wave tile into out.
    // C/D layout: VGPR i -> M = i (lanes 0-15) / i+8 (lanes 16-31); N = lane%16.
#pragma unroll
    for (int j = 0; j < 4; ++j) {
        const int col = colBase + j * 16 + m;
        float* __restrict__ op0 = out + (long long)(rowBase + half * 8) * DDIM + col;
        float* __restrict__ op1 = op0 + 16 * DDIM;
#pragma unroll
        for (int i = 0; i < 8; ++i) {
            op0[(long long)i * DDIM] += c[j][0][i];
            op1[(long long)i * DDIM] += c[j][1][i];
        }
    }
}

// ---------------------------------------------------------------------------
// h = acc + x + b ; LayerNorm(h) * g + beta ; PReLU.  One wave32 per row.
// ---------------------------------------------------------------------------
__global__ void rgcn_ln_prelu_kernel(const float* __restrict__ acc,
                                     const float* __restrict__ x,
                                     const float* __restrict__ bias,
                                     const float* __restrict__ gamma,
                                     const float* __restrict__ beta,
                                     const float* __restrict__ prelu_a,
                                     float* __restrict__ outp) {
    const int lane = threadIdx.x & 31;
    const int wrow = threadIdx.x >> 5;
    const long long row = (long long)blockIdx.x * 8 + wrow;
    if (row >= N_NODES) return;

    const float* __restrict__ ap = acc + row * DDIM;
    const float* __restrict__ xp = x   + row * DDIM;

    float h[8];
    float sum = 0.0f;
#pragma unroll
    for (int j = 0; j < 8; ++j) {
        const int cc = lane + j * 32;
        h[j] = ap[cc] + xp[cc] + bias[cc];
        sum += h[j];
    }
#pragma unroll
    for (int off = 16; off > 0; off >>= 1) sum += __shfl_xor(sum, off, 32);
    const float mu = sum * (1.0f / 256.0f);

    float vs = 0.0f;
#pragma unroll
    for (int j = 0; j < 8; ++j) {
        const float d = h[j] - mu;
        vs += d * d;
    }
#pragma unroll
    for (int off = 16; off > 0; off >>= 1) vs += __shfl_xor(vs, off, 32);
    const float inv = rsqrtf(vs * (1.0f / 256.0f) + 1e-5f);

    const float aP = prelu_a[0];
    float* __restrict__ op = outp + row * DDIM;
#pragma unroll
    for (int j = 0; j < 8; ++j) {
        const int cc = lane + j * 32;
        const float nm = (h[j] - mu) * inv * gamma[cc] + beta[cc];
        op[cc] = (nm >= 0.0f) ? nm : aP * nm;
    }
}

// ---------------------------------------------------------------------------
extern "C" void kernel_launch(void* const* d_in, const int* in_sizes, int n_in,
                              void* d_out, int out_size, void* d_ws, size_t ws_size,
                              hipStream_t stream) {
    (void)in_sizes; (void)n_in; (void)out_size; (void)ws_size;

    const float* x        = (const float*)d_in[0];
    const int*   edges    = (const int*)d_in[1];    // (R, 2, E)
    const float* W_rel    = (const float*)d_in[2];  // (R, D, D)
    const float* W_self_w = (const float*)d_in[3];  // (D, D)
    const float* W_self_b = (const float*)d_in[4];  // (D,)
    const float* ln_g     = (const float*)d_in[5];
    const float* ln_b     = (const float*)d_in[6];
    const float* prelu_a  = (const float*)d_in[7];

    float* A   = (float*)d_ws;                       // N*D accumulator
    float* deg = A + (size_t)N_NODES * DDIM;         // N degrees
    float* out = (float*)d_out;                      // N*D fp32 accumulator -> final

    (void)hipMemsetAsync(out, 0, (size_t)N_NODES * DDIM * sizeof(float), stream);

    const dim3 blkScatter(256);
    const dim3 gridScatter(((size_t)E_EDGES * 64) / 256);   // 50000
    const dim3 blkGemm(128);                                // 4 waves
    const dim3 gridGemm(N_NODES / 32);                      // 3125
    const dim3 blkLN(256);
    const dim3 gridLN(N_NODES / 8);                         // 12500
    const size_t smemGemm = 2 * BUFELEM * sizeof(float);    // 64 KB

    for (int r = 0; r < R_REL; ++r) {
        (void)hipMemsetAsync(A, 0, ((size_t)N_NODES * DDIM + N_NODES) * sizeof(float), stream);
        const int* src = edges + (size_t)r * 2 * E_EDGES;
        const int* dst = src + E_EDGES;
        rgcn_scatter_kernel<<<gridScatter, blkScatter, 0, stream>>>(x, src, dst, A, deg);
        rgcn_gemm_acc_kernel<<<gridGemm, blkGemm, smemGemm, stream>>>(
            A, deg, W_rel + (size_t)r * DDIM * DDIM, out);
    }

    // Self-loop term: out += x @ W_self  (no degree scaling).
    rgcn_gemm_acc_kernel<<<gridGemm, blkGemm, smemGemm, stream>>>(
        x, nullptr, W_self_w, out);

    // Bias + residual + LayerNorm + PReLU (in-place on d_out).
    rgcn_ln_prelu_kernel<<<gridLN, blkLN, 0, stream>>>(
        out, x, W_self_b, ln_g, ln_b, prelu_a, out);
}